// LayoutModel_2327872274937
// MI455X (gfx1250) — compile-verified
//
#include <hip/hip_runtime.h>
#include <hip/hip_bf16.h>

typedef __attribute__((ext_vector_type(2))) float v2f;
typedef __attribute__((ext_vector_type(8))) float v8f;

#define B_ 32
#define T_ 2048
#define E_ 128
#define L_ 17

// ---------------------------------------------------------------------------
// Kernel 1: fused embedding gather + emissions GEMM via V_WMMA_F32_16X16X4_F32
//   em[r][l] = sum_k emb[text[r]][k] * W5[k][l] + b5[l],  r in [0, B*T)
// One wave32 computes a 16-row x 17-col tile (two 16x16 WMMA accumulators,
// cols 17..31 of the padded B tile are zero). 2 waves / 64-thread block.
// W is stored TRANSPOSED in LDS so each B fragment is a contiguous 8-byte
// pair -> single ds_load_b64 per fragment, no VGPR repacking.
// ---------------------------------------------------------------------------
__global__ __launch_bounds__(64) void emissions_wmma_kernel(
    const int* __restrict__ text,      // [B*T]
    const float* __restrict__ emb,     // [V,128]
    const float* __restrict__ W5,      // [128,17]
    const float* __restrict__ b5,      // [17]
    float* __restrict__ em)            // [B*T,17]
{
    __shared__ __align__(16) float sWt[32][132];      // W5^T padded to 32 rows
    __shared__ __align__(16) float sA[2][16][132];    // per-wave 16x128 A tile

    const int tid  = threadIdx.x;
    const int wave = tid >> 5;
    const int lane = tid & 31;

    // Stage transposed, padded W5 (32 x 128) into LDS
    for (int idx = tid; idx < 32 * E_; idx += 64) {
        const int n = idx >> 7;        // 0..31  (output column)
        const int k = idx & 127;       // 0..127 (reduction index)
        sWt[n][k] = (n < L_) ? W5[k * L_ + n] : 0.0f;
    }

    const int rowbase = blockIdx.x * 32 + wave * 16;

    // Gather 16 rows of embeddings (2 lanes per row, 64 floats each)
    {
        const int row  = lane >> 1;
        const int half = lane & 1;
        const int tok  = text[rowbase + row];
        const float4* src = (const float4*)(emb + (size_t)tok * E_ + half * 64);
        float* dst = &sA[wave][row][half * 64];
        #pragma unroll
        for (int it = 0; it < 16; ++it) {
            float4 v = src[it];
            *(float4*)&dst[it * 4] = v;
        }
    }
    __syncthreads();

    // A fragment (16x4 f32): lanes 0-15 hold K={k,k+1}, lanes 16-31 K={k+2,k+3}
    // B fragment (4x16 f32): same K split, N = lane&15
    v8f acc0 = {};  // cols 0..15
    v8f acc1 = {};  // cols 16..31 (only col 16 stored)
    const int n    = lane & 15;          // doubles as M for A, N for B
    const int koff = (lane >> 4) * 2;

    #pragma unroll
    for (int ks = 0; ks < 32; ++ks) {
        const int k = ks * 4 + koff;     // even -> 8B-aligned LDS pairs
        v2f a  = *(const v2f*)&sA[wave][n][k];
        v2f b0 = *(const v2f*)&sWt[n][k];
        v2f b1 = *(const v2f*)&sWt[16 + n][k];
        acc0 = __builtin_amdgcn_wmma_f32_16x16x4_f32(
                   false, a, false, b0, (short)0, acc0, false, false);
        acc1 = __builtin_amdgcn_wmma_f32_16x16x4_f32(
                   false, a, false, b1, (short)0, acc1, false, false);
    }

    // C/D layout: VGPR v -> M = v + (lane>=16 ? 8 : 0), N = lane&15
    const int mofs = (lane >> 4) * 8;
    const float bn  = b5[n];
    const float b16 = b5[16];
    #pragma unroll
    for (int v = 0; v < 8; ++v) {
        const size_t row = (size_t)(rowbase + mofs + v);
        em[row * L_ + n] = acc0[v] + bn;
        if (n == 0) em[row * L_ + 16] = acc1[v] + b16;
    }
}

// ---------------------------------------------------------------------------
// Kernel 2: Viterbi scan + backtrace. One wave32 per batch chain.
// Lane j keeps alpha[j]; alpha[i] broadcast via wave32 shuffles.
// Backpointers stored in LDS (2047 x 17 bytes) for fast backtrace.
// ---------------------------------------------------------------------------
__global__ __launch_bounds__(32) void viterbi_kernel(
    const float* __restrict__ em,     // [B,T,17]
    const float* __restrict__ trans,  // [17,17]
    int* __restrict__ tags,           // [B,T]
    float* __restrict__ score,        // [B]
    int* __restrict__ lens)           // [B]
{
    __shared__ unsigned char sbp[(T_ - 1) * L_];

    const int b    = blockIdx.x;
    const int lane = threadIdx.x;
    const bool valid = lane < L_;
    const int j = valid ? lane : 0;

    // trans column j in registers (fully unrolled -> stays in VGPRs)
    float tcol[L_];
    #pragma unroll
    for (int i = 0; i < L_; ++i) tcol[i] = trans[i * L_ + j];

    const float* emB = em + (size_t)b * T_ * L_;
    float alpha     = emB[j];        // t = 0 emission
    float emit_next = emB[L_ + j];   // prefetch t = 1

    for (int t = 1; t < T_; ++t) {
        const float emit_cur = emit_next;
        if (t + 1 < T_) emit_next = emB[(size_t)(t + 1) * L_ + j];

        float best = -3.402823466e38f;
        int bi = 0;
        #pragma unroll
        for (int i = 0; i < L_; ++i) {
            const float ai = __shfl(alpha, i, 32);
            const float s = ai + tcol[i];
            if (s > best) { best = s; bi = i; }   // first-max tie break
        }
        alpha = best + emit_cur;
        if (valid) sbp[(t - 1) * L_ + lane] = (unsigned char)bi;
    }
    __syncthreads();

    // argmax over lanes 0..16 (first max wins)
    float bestv = -3.402823466e38f;
    int bestj = 0;
    #pragma unroll
    for (int i = 0; i < L_; ++i) {
        const float v = __shfl(alpha, i, 32);
        if (v > bestv) { bestv = v; bestj = i; }
    }

    if (lane == 0) {
        score[b] = bestv;
        lens[b]  = T_;
        int* tb = tags + (size_t)b * T_;
        int tag = bestj;
        tb[T_ - 1] = tag;
        for (int t = T_ - 2; t >= 0; --t) {
            tag = sbp[t * L_ + tag];
            tb[t] = tag;
        }
    }
}

// ---------------------------------------------------------------------------
extern "C" void kernel_launch(void* const* d_in, const int* in_sizes, int n_in,
                              void* d_out, int out_size, void* d_ws, size_t ws_size,
                              hipStream_t stream) {
    (void)in_sizes; (void)n_in; (void)out_size; (void)ws_size;

    const int*   text  = (const int*)d_in[0];    // [B,T]
    const float* emb   = (const float*)d_in[1];  // [V,E]
    const float* W5    = (const float*)d_in[2];  // [E,L]
    const float* b5    = (const float*)d_in[3];  // [L]
    const float* trans = (const float*)d_in[4];  // [L,L]

    float* em = (float*)d_ws;                    // [B*T, 17] scratch (~4.5 MB)

    int*   tags  = (int*)d_out;                          // [B,T] int32 bits
    float* score = (float*)d_out + (size_t)B_ * T_;      // [B]
    int*   lens  = (int*)((float*)d_out + (size_t)B_ * T_ + B_);  // [B]

    emissions_wmma_kernel<<<(B_ * T_) / 32, 64, 0, stream>>>(text, emb, W5, b5, em);
    viterbi_kernel<<<B_, 32, 0, stream>>>(em, trans, tags, score, lens);
}